// UpFilter2d_31748398252691
// MI455X (gfx1250) — compile-verified
//
#include <hip/hip_runtime.h>

// UpFilter2d (zero-insert upsample x2 + depthwise 5x5 Pascal filter) on gfx1250.
// Memory-bound: 335 MB traffic -> ~14.4us floor @ 23.3 TB/s. Per 16x16 output
// tile: Out = Wv * X * Wh via V_WMMA_F32_16X16X4_F32 (exact f32 weights).
// Round-5 changes vs round-4 (652 instr / 8KB out):
//  * K=12 instead of 16: the polyphase weight matrix has only 10 nonzero rows,
//    and step s covers K=4s+2*half+{0,1}; the s=3 fragment is identically zero
//    (round-4 disasm proved it: v_mov_b64 v[6:7],0). 3 WMMA steps per stage
//    -> 48 WMMAs/wave, 16 fewer LDS loads, bit-identical results.
//  * patch narrowed to 10 float4 chunks/row (max read col is 38 < 40):
//    loader is 8 iterations instead of 9.

typedef __attribute__((ext_vector_type(2))) float v2f;
typedef __attribute__((ext_vector_type(4))) float v4f;
typedef __attribute__((ext_vector_type(8))) float v8f;

#define HIN 128
#define WIN 128
#define HOUT 256
#define WOUT 256
#define PSTRIDE 44   // patch row stride: 44*lm mod 64 distinct (order 16);
                     // +2-dword half-wave collision needs 22d==1 mod 32 -> none
#define CHUNKS 10    // float4 chunks loaded per patch row (40 cols, reads <=38)
#define TSTRIDE 20   // transposed-T row stride: b128 stores 2 dwords/bank,
                     // b64 loads conflict-free, rows 16B aligned
#define DEADZ 96     // slack for stray chunk stores (g in [240,256) -> <=1139)
#define WREG (24 * PSTRIDE + DEADZ + 16 * TSTRIDE)  // 1056+96+320 = 1472 floats

__global__ __launch_bounds__(256) void upfilter2d_wmma(const float* __restrict__ xin,
                                                       float* __restrict__ out) {
  __shared__ __align__(16) float lds[8 * WREG];

  const int lane  = threadIdx.x & 31;
  const int wslot = threadIdx.x >> 5;
  const int half  = lane >> 4;   // selects K pair within a WMMA step
  const int lm    = lane & 15;   // M for A-fragments, N for B/C/D fragments

  float* lX = &lds[wslot * WREG];
  float* lT = lX + (24 * PSTRIDE + DEADZ);

  // One wave per 32x64 output block: 1024 planes * 32 blocks = 32768 waves.
  const int wid   = blockIdx.x * 8 + wslot;
  const int plane = wid >> 5;
  const int tidx  = wid & 31;
  const int i0    = (tidx >> 2) << 5;   // output block row origin (8 x 32)
  const int j0    = (tidx & 3) << 6;    // output block col origin (4 x 64)
  const int m0    = (i0 >> 1) - 1;      // patch row origin (>= -1)
  const int n0a   = (j0 >> 1) - 4;      // patch col origin, 4-aligned (>= -4)

  const size_t ibase = (size_t)plane * (HIN * WIN);
  const size_t obase = (size_t)plane * (HOUT * WOUT);

  // ---- Stage 0: 24 rows x 10 float4 chunks (cols n0a..n0a+39) into LDS. ----
  // Chunk g = k*32+lane -> (r, c4) = (g/10, g%10); g in [240,256) lands in the
  // dead zone (never read). Plane-clamped addresses keep every load in-bounds;
  // masked-out chunks store 0 (the conv's zero padding) so all patch cells are
  // finite; out-of-window cells are killed by zero rows/cols of Wh/Wv.
  {
    int r  = lane / 10;
    int c4 = lane - 10 * r;
#pragma unroll
    for (int k = 0; k < 8; ++k) {
      int xr = m0 + r;
      int xc = n0a + 4 * c4;
      bool ok = ((unsigned)xr < (unsigned)HIN) && ((unsigned)xc < (unsigned)WIN);
      int xrc = xr < 0 ? 0 : (xr > HIN - 1 ? HIN - 1 : xr);
      int xcc = xc < 0 ? 0 : (xc > WIN - 4 ? WIN - 4 : xc);
      v4f v = *(const v4f*)(xin + ibase + ((size_t)xrc << 7) + xcc);
      v4f zero = {0.f, 0.f, 0.f, 0.f};
      *(v4f*)&lX[r * PSTRIDE + 4 * c4] = ok ? v : zero;
      // advance chunk index by 32: 32 = 3*10 + 2
      r += 3;
      c4 += 2;
      if (c4 >= 10) { c4 -= 10; r += 1; }
    }
  }

  // ---- Coefficient fragments (branch-free; shared by all 8 sub-tiles). ----
  // Weight(input r -> output q) = pascal[2r-q]/16, pascal={1,4,6,4,1}, else 0.
  // wc0[s] = W(kk,lm), wc1[s] = W(kk+1,lm), kk = 4s+2*half:
  //   d0 = 2*kk-lm = dbase+8s; wc1 uses d0+2 -> |d-2| = |d0|.
  // s=3 is identically zero (d0 >= 9) -> only 3 steps needed.
  const int dbase = 4 * half - lm;
  float wc0[3], wc1[3];
#pragma unroll
  for (int s = 0; s < 3; ++s) {
    int d0 = dbase + 8 * s;
    int e0 = abs(d0 - 2);
    int e1 = abs(d0);
    wc0[s] = e0 == 0 ? 0.375f : (e0 == 1 ? 0.25f : (e0 == 2 ? 0.0625f : 0.0f));
    wc1[s] = e1 == 0 ? 0.375f : (e1 == 1 ? 0.25f : (e1 == 2 ? 0.0625f : 0.0f));
  }

  // Hoisted output base: all 64 stores below use immediate offsets.
  float* po = out + obase + (size_t)(i0 + 8 * half) * WOUT + (j0 + lm);

  // ---- 2x4 sub-tiles of 16x16 outputs each. ----
#pragma unroll
  for (int a = 0; a < 2; ++a) {
#pragma unroll
    for (int b = 0; b < 4; ++b) {
      // Window (ri,rj) -> patch[8a+ri][8b+3+rj]; max col read = 24+3+11 = 38
      const int rbase = (8 * a + lm) * PSTRIDE + 8 * b + 3;

      // Stage 1: T = Xwin * Wh (horizontal polyphase), K=12 in 3 steps.
      v8f t = {0.f, 0.f, 0.f, 0.f, 0.f, 0.f, 0.f, 0.f};
#pragma unroll
      for (int s = 0; s < 3; ++s) {
        int kk = 4 * s + 2 * half;
        v2f av = {lX[rbase + kk], lX[rbase + kk + 1]};  // -> ds_load_2addr
        v2f bv = {wc0[s], wc1[s]};
        t = __builtin_amdgcn_wmma_f32_16x16x4_f32(false, av, false, bv, 0, t,
                                                  false, false);
      }

      // Bounce T transposed: lane's 8 accumulators contiguous -> 2x b128.
      {
        v4f lo = {t[0], t[1], t[2], t[3]};
        v4f hi = {t[4], t[5], t[6], t[7]};
        v4f* tp = (v4f*)&lT[lm * TSTRIDE + 8 * half];
        tp[0] = lo;
        tp[1] = hi;
      }

      // Stage 2: Out = Wv * T (vertical polyphase), K=12. B pair -> b64.
      v8f o = {0.f, 0.f, 0.f, 0.f, 0.f, 0.f, 0.f, 0.f};
#pragma unroll
      for (int s = 0; s < 3; ++s) {
        int kk = 4 * s + 2 * half;
        v2f av = {wc0[s], wc1[s]};
        v2f bv = *(const v2f*)&lT[lm * TSTRIDE + kk];   // -> ds_load_b64
        o = __builtin_amdgcn_wmma_f32_16x16x4_f32(false, av, false, bv, 0, o,
                                                  false, false);
      }

      // Store D: VGPR v -> row (16a + v + 8*half), col (16b + lm).
#pragma unroll
      for (int v = 0; v < 8; ++v)
        po[(16 * a + v) * WOUT + 16 * b] = o[v];
    }
  }
}

extern "C" void kernel_launch(void* const* d_in, const int* in_sizes, int n_in,
                              void* d_out, int out_size, void* d_ws, size_t ws_size,
                              hipStream_t stream) {
  const float* x = (const float*)d_in[0];   // (8,128,128,128) f32
  // d_in[1] = Pascal kernel (constant, baked into the coefficient table);
  // d_in[2] = scale=2.
  float* out = (float*)d_out;               // (8,128,256,256) f32

  // 32768 waves / 8 waves per 256-thread block
  dim3 grid(4096), block(256);
  upfilter2d_wmma<<<grid, block, 0, stream>>>(x, out);
}